// CLDiceLoss_37263136260423
// MI455X (gfx1250) — compile-verified
//
#include <hip/hip_runtime.h>
#include <stdint.h>

// clDice loss, bit-packed morphological thinning for gfx1250 (MI455X).
// One workgroup per image; each thread owns one 512-pixel row as 16 u32 words
// in registers. 40 thinning iterations run entirely in registers + LDS with
// 2 workgroup barriers each. Final 16-wave x 4-counter reduction uses
// v_wmma_f32_16x16x4_f32.

typedef __attribute__((ext_vector_type(2))) float v2f;
typedef __attribute__((ext_vector_type(8))) float v8f;

#define B_IMG 32
#define H_IMG 512
#define W_IMG 512
#define NW    16   // u32 words per row
#define NWP   17   // padded LDS row stride (gcd(17,64)=1 -> conflict free)
#define KITER 40
#define EPS_F 1e-6f
// EPS is added elementwise before the numerator sums: 512*512*1e-6
#define EPS_SUM 0.262144f

__global__ __launch_bounds__(H_IMG) void cldice_skel_kernel(
    const float* __restrict__ pred,
    const float* __restrict__ gtm,
    const float* __restrict__ gts,
    float* __restrict__ cl_out)
{
  extern __shared__ uint32_t smem[];
  uint32_t* lA = smem;                          // [H_IMG][NWP]
  uint32_t* lB = smem + H_IMG * NWP;            // [H_IMG][NWP]
  float*    P  = (float*)(smem + 2 * H_IMG * NWP);  // [16 waves][4 counters]

  const int r = threadIdx.x;          // row index within the image
  const int b = blockIdx.x;           // image index
  const size_t rowoff = ((size_t)b * H_IMG + (size_t)r) * W_IMG;

  const float4* p4 = (const float4*)(pred + rowoff);
  const float4* g4 = (const float4*)(gtm + rowoff);
  const float4* s4 = (const float4*)(gts + rowoff);

  __builtin_prefetch(pred + rowoff, 0, 0);   // global_prefetch_b8
  __builtin_prefetch(gtm + rowoff, 0, 0);
  __builtin_prefetch(gts + rowoff, 0, 0);

  // ---- Load + threshold + bit-pack; t_rec popcounts use pre-skeleton pred.
  uint32_t x[NW];    // binary_pred bits (becomes pred_skel after thinning)
  uint32_t gt[NW];   // binary_gt bits
  int rec_num = 0, rec_den = 0;

#pragma unroll 4
  for (int i = 0; i < NW; ++i) {
    uint32_t xw = 0u, gw = 0u, sw = 0u;
#pragma unroll
    for (int j = 0; j < 8; ++j) {
      const float4 pv = p4[i * 8 + j];
      const float4 gv = g4[i * 8 + j];
      const float4 sv = s4[i * 8 + j];
      const int base = j * 4;
      xw |= (pv.x >= 0.5f ? 1u : 0u) << (base + 0);
      xw |= (pv.y >= 0.5f ? 1u : 0u) << (base + 1);
      xw |= (pv.z >= 0.5f ? 1u : 0u) << (base + 2);
      xw |= (pv.w >= 0.5f ? 1u : 0u) << (base + 3);
      gw |= (gv.x != 0.0f ? 1u : 0u) << (base + 0);
      gw |= (gv.y != 0.0f ? 1u : 0u) << (base + 1);
      gw |= (gv.z != 0.0f ? 1u : 0u) << (base + 2);
      gw |= (gv.w != 0.0f ? 1u : 0u) << (base + 3);
      sw |= (sv.x != 0.0f ? 1u : 0u) << (base + 0);
      sw |= (sv.y != 0.0f ? 1u : 0u) << (base + 1);
      sw |= (sv.z != 0.0f ? 1u : 0u) << (base + 2);
      sw |= (sv.w != 0.0f ? 1u : 0u) << (base + 3);
    }
    x[i] = xw;
    gt[i] = gw;
    rec_num += __popc(sw & xw);   // sum(gt_skel * binary_pred)
    rec_den += __popc(sw);        // sum(gt_skel)
  }

  // ---- 40 iterations of soft_skeletonize on binary data:
  //   e = erode3x3(x)  (AND, OOB=1)   [= min_pool]
  //   d = dilate3x3(e) (OR,  OOB=0)   [= maxpool(min_pool)]
  //   x = x & ~(d & ~e)               [= relu(x - relu(d - e))]
#pragma unroll 1
  for (int it = 0; it < KITER; ++it) {
    // Phase 1: horizontal AND of x within the row -> lA[r][*]
    uint32_t h[NW];
#pragma unroll
    for (int i = 0; i < NW; ++i) {
      const uint32_t lw = (i > 0)      ? x[i - 1] : 0xFFFFFFFFu; // OOB col = 1
      const uint32_t rw = (i < NW - 1) ? x[i + 1] : 0xFFFFFFFFu;
      h[i] = x[i] & ((x[i] << 1) | (lw >> 31)) & ((x[i] >> 1) | (rw << 31));
    }
#pragma unroll
    for (int i = 0; i < NW; ++i) lA[r * NWP + i] = h[i];
    __syncthreads();   // lA published; also guards lB reuse from prev iter

    // Phase 2: vertical AND -> e; then horizontal OR of e -> lB[r][*]
    uint32_t e[NW];
#pragma unroll
    for (int i = 0; i < NW; ++i) {
      const uint32_t up = (r > 0)         ? lA[(r - 1) * NWP + i] : 0xFFFFFFFFu;
      const uint32_t dn = (r < H_IMG - 1) ? lA[(r + 1) * NWP + i] : 0xFFFFFFFFu;
      e[i] = h[i] & up & dn;
    }
    uint32_t ho[NW];
#pragma unroll
    for (int i = 0; i < NW; ++i) {
      const uint32_t lw = (i > 0)      ? e[i - 1] : 0u;  // OOB col = 0
      const uint32_t rw = (i < NW - 1) ? e[i + 1] : 0u;
      ho[i] = e[i] | (e[i] << 1) | (lw >> 31) | (e[i] >> 1) | (rw << 31);
    }
#pragma unroll
    for (int i = 0; i < NW; ++i) lB[r * NWP + i] = ho[i];
    __syncthreads();   // lB published; all lA reads done

    // Phase 3: vertical OR -> d; update x in registers
#pragma unroll
    for (int i = 0; i < NW; ++i) {
      const uint32_t up = (r > 0)         ? lB[(r - 1) * NWP + i] : 0u;
      const uint32_t dn = (r < H_IMG - 1) ? lB[(r + 1) * NWP + i] : 0u;
      const uint32_t d = ho[i] | up | dn;
      x[i] &= ~(d & ~e[i]);
    }
    // next iteration's first barrier protects lB before it is rewritten
  }

  // ---- t_prec popcounts on the skeleton
  int prec_num = 0, prec_den = 0;
#pragma unroll
  for (int i = 0; i < NW; ++i) {
    prec_num += __popc(x[i] & gt[i]);   // sum(pred_skel * binary_gt)
    prec_den += __popc(x[i]);           // sum(pred_skel)
  }

  // ---- per-wave butterfly reduction (wave32)
#pragma unroll
  for (int off = 16; off > 0; off >>= 1) {
    prec_num += __shfl_xor(prec_num, off);
    prec_den += __shfl_xor(prec_den, off);
    rec_num  += __shfl_xor(rec_num, off);
    rec_den  += __shfl_xor(rec_den, off);
  }
  const int wave = threadIdx.x >> 5;
  if ((threadIdx.x & 31) == 0) {
    P[wave * 4 + 0] = (float)prec_num;
    P[wave * 4 + 1] = (float)prec_den;
    P[wave * 4 + 2] = (float)rec_num;
    P[wave * 4 + 3] = (float)rec_den;
  }
  __syncthreads();

  // ---- cross-wave reduction of the 16x4 partial matrix via WMMA:
  // A[m][k] = P[4k + (m&3)][m>>2], B = ones(4x16), D row-sums give totals.
  if (threadIdx.x < 32) {
    const int ln = threadIdx.x;
    const int m  = ln & 15;
    const int k0 = (ln < 16) ? 0 : 2;     // ISA A layout: VGPR0 holds K=0/K=2
    v2f a;
    a.x = P[(4 * k0 + (m & 3)) * 4 + (m >> 2)];
    a.y = P[(4 * (k0 + 1) + (m & 3)) * 4 + (m >> 2)];
    v2f bm;
    bm.x = 1.0f;
    bm.y = 1.0f;
    v8f c = {};
    v8f d = __builtin_amdgcn_wmma_f32_16x16x4_f32(
        /*neg_a=*/false, a, /*neg_b=*/false, bm,
        /*c_mod=*/(short)0, c, /*reuse_a=*/false, /*reuse_b=*/false);
    // lanes 0-15: d[v] = rowsum[v];  lanes 16-31: d[v] = rowsum[v+8]
    const float s01 = d[0] + d[1] + d[2] + d[3];  // totals[0] / totals[2]
    const float s23 = d[4] + d[5] + d[6] + d[7];  // totals[1] / totals[3]
    const float prec_num_f = __shfl(s01, 0);
    const float prec_den_f = __shfl(s23, 0);
    const float rec_num_f  = __shfl(s01, 16);
    const float rec_den_f  = __shfl(s23, 16);
    if (ln == 0) {
      const float t_prec = (prec_num_f + EPS_SUM) / (prec_den_f + EPS_F);
      const float t_rec  = (rec_num_f  + EPS_SUM) / (rec_den_f  + EPS_F);
      cl_out[b] = 2.0f * (t_prec * t_rec / (t_prec + t_rec));
    }
  }
}

__global__ void cldice_finalize_kernel(const float* __restrict__ cl,
                                       float* __restrict__ out)
{
  if (threadIdx.x == 0 && blockIdx.x == 0) {
    float s = 0.0f;
#pragma unroll
    for (int i = 0; i < B_IMG; ++i) s += cl[i];  // fixed order -> deterministic
    out[0] = 1.0f - s * (1.0f / (float)B_IMG);
  }
}

extern "C" void kernel_launch(void* const* d_in, const int* in_sizes, int n_in,
                              void* d_out, int out_size, void* d_ws, size_t ws_size,
                              hipStream_t stream)
{
  const float* pred = (const float*)d_in[0];   // pred_binary_mask [32,512,512] f32
  const float* gtm  = (const float*)d_in[1];   // gt_mask          [32,512,512] f32
  const float* gts  = (const float*)d_in[2];   // gt_skel          [32,512,512] f32
  float* cl = (float*)d_ws;                    // 32 per-image cldice values

  const size_t shmem = (size_t)(2 * H_IMG * NWP) * sizeof(uint32_t)
                     + (size_t)(16 * 4) * sizeof(float);   // ~70 KB (< 320 KB/WGP)

  cldice_skel_kernel<<<dim3(B_IMG), dim3(H_IMG), shmem, stream>>>(pred, gtm, gts, cl);
  cldice_finalize_kernel<<<dim3(1), dim3(32), 0, stream>>>(cl, (float*)d_out);
}